// BerryAMXAttention_56126632624500
// MI455X (gfx1250) — compile-verified
//
#include <hip/hip_runtime.h>
#include <math.h>

// ---------------------------------------------------------------------------
// Quaternion attention (BerryAMXAttention), MI455X / gfx1250, fp32.
//   B=2, L=384, E=512, H=8, head_dim=64, atoms c=16.
// Stage 1: q/k/v = x @ W^T          -> f32 WMMA 16x16x4 GEMM
// Stage 2: fused pairwise hamilton/norm/gate/aggregate (VALU, wave32)
// Stage 3: out = ctx @ Wo^T         -> same WMMA GEMM
// ---------------------------------------------------------------------------

typedef __attribute__((ext_vector_type(2))) float v2f;
typedef __attribute__((ext_vector_type(8))) float v8f;

#define EDIM 512
#define LDIM 384
#define ROWS 768   // B*L

// Y[r, n] = sum_e X[r, e] * W[n, e]   (Y = X @ W^T), all 512-wide rows.
// One wave -> 32x32 output tile = 2x2 WMMA f32 16x16 fragments.
__global__ __launch_bounds__(256)
void quat_proj_gemm(const float* __restrict__ X,
                    const float* __restrict__ W,
                    float* __restrict__ Y) {
    const int wave   = (blockIdx.x * blockDim.x + threadIdx.x) >> 5;
    const int tilesN = EDIM / 32;                 // 16
    const int m0 = (wave / tilesN) * 32;
    const int n0 = (wave % tilesN) * 32;
    if (m0 >= ROWS) return;

    const int lane = threadIdx.x & 31;
    const int r    = lane & 15;                   // row-in-tile (A) / col-in-tile (B)
    const int hh   = lane >> 4;                   // lane half selects K pair

    // A frag (16x4 f32): lane holds M=r, K = 2*hh + {0,1}  -> one float2 load.
    // B frag (4x16 f32): lane holds N=r, K = 2*hh + {0,1}  -> identical math.
    const float* xa0 = X + (size_t)(m0 + r) * EDIM + 2 * hh;
    const float* xa1 = xa0 + 16 * EDIM;
    const float* wb0 = W + (size_t)(n0 + r) * EDIM + 2 * hh;
    const float* wb1 = wb0 + 16 * EDIM;

    v8f c00 = {}, c01 = {}, c10 = {}, c11 = {};

#pragma unroll 4
    for (int k = 0; k < EDIM; k += 4) {
        v2f a0 = *(const v2f*)(xa0 + k);
        v2f a1 = *(const v2f*)(xa1 + k);
        v2f b0 = *(const v2f*)(wb0 + k);
        v2f b1 = *(const v2f*)(wb1 + k);
        c00 = __builtin_amdgcn_wmma_f32_16x16x4_f32(false, a0, false, b0,
                                                    (short)0, c00, false, false);
        c01 = __builtin_amdgcn_wmma_f32_16x16x4_f32(false, a0, false, b1,
                                                    (short)0, c01, false, false);
        c10 = __builtin_amdgcn_wmma_f32_16x16x4_f32(false, a1, false, b0,
                                                    (short)0, c10, false, false);
        c11 = __builtin_amdgcn_wmma_f32_16x16x4_f32(false, a1, false, b1,
                                                    (short)0, c11, false, false);
    }

    // C/D layout: VGPR v, lanes 0-15 -> M=v, lanes 16-31 -> M=v+8; N = lane&15.
    const int col = n0 + r;
#pragma unroll
    for (int v = 0; v < 8; ++v) {
        const int row = m0 + v + 8 * hh;
        Y[(size_t)row * EDIM + col]              = c00[v];
        Y[(size_t)row * EDIM + col + 16]         = c01[v];
        Y[(size_t)(row + 16) * EDIM + col]       = c10[v];
        Y[(size_t)(row + 16) * EDIM + col + 16]  = c11[v];
    }
}

__device__ __forceinline__ float sigmoidf_fast(float x) {
    return 1.0f / (1.0f + __expf(-x));
}

// One wave per (bh, i). Lane = (half: which j of the pair, atom 0..15).
// Loops over key positions j two at a time; ctx accumulated in registers.
__global__ __launch_bounds__(256)
void quat_attn(const float* __restrict__ q,
               const float* __restrict__ k,
               const float* __restrict__ v,
               const float* __restrict__ dde_w,
               const float* __restrict__ dde_b,
               float* __restrict__ ctx) {
    const int wave = (blockIdx.x * blockDim.x + threadIdx.x) >> 5;  // 0..6143
    const int bh = wave / LDIM;            // 0..15
    const int i  = wave % LDIM;
    const int b  = bh >> 3;
    const int h  = bh & 7;

    const int lane = threadIdx.x & 31;
    const int atom = lane & 15;
    const int half = lane >> 4;

    // Gate parameters (uniform -> scalar loads).
    float dw[16], db[4];
#pragma unroll
    for (int p = 0; p < 16; ++p) dw[p] = dde_w[p];
#pragma unroll
    for (int p = 0; p < 4; ++p) db[p] = dde_b[p];

    const size_t headOff = (size_t)h * 64 + (size_t)atom * 4;
    const float* qp = q + ((size_t)(b * LDIM + i) * EDIM) + headOff;
    const float qw = qp[0], qx = qp[1], qy = qp[2], qz = qp[3];

    float aw = 0.f, ax = 0.f, ay = 0.f, az = 0.f;

    const float* kbase = k + (size_t)b * LDIM * EDIM + headOff;
    const float* vbase = v + (size_t)b * LDIM * EDIM + headOff;

    for (int j0 = 0; j0 < LDIM; j0 += 2) {
        const int j = j0 + half;

        // Prefetch the k/v rows ~16 iterations ahead (lands in ws; speculative).
        __builtin_prefetch(kbase + (size_t)(j + 32) * EDIM, 0, 1);
        __builtin_prefetch(vbase + (size_t)(j + 32) * EDIM, 0, 1);

        const float4 kq = *(const float4*)(kbase + (size_t)j * EDIM);
        const float kw = kq.x, kx = kq.y, ky = kq.z, kz = kq.w;

        // interference = hamilton(q_i, k_j)
        float iw = qw * kw - qx * kx - qy * ky - qz * kz;
        float ix = qw * kx + qx * kw + qy * kz - qz * ky;
        float iy = qw * ky - qx * kz + qy * kw + qz * kx;
        float iz = qw * kz + qx * ky - qy * kx + qz * kw;

        const float n2  = iw * iw + ix * ix + iy * iy + iz * iz;
        const float inv = 1.0f / (__fsqrt_rn(n2) + 1e-6f);
        float sw = iw * inv, sx = ix * inv, sy = iy * inv, sz = iz * inv;

        // mean over 16 atoms (lanes within this half-wave)
        float mw = sw, mx = sx, my = sy, mz = sz;
#pragma unroll
        for (int m = 1; m < 16; m <<= 1) {
            mw += __shfl_xor(mw, m);
            mx += __shfl_xor(mx, m);
            my += __shfl_xor(my, m);
            mz += __shfl_xor(mz, m);
        }
        mw *= 0.0625f; mx *= 0.0625f; my *= 0.0625f; mz *= 0.0625f;

        // gate[p] = sigmoid(sum_q mean[q] * dde_w[p][q] + dde_b[p])
        const float g0 = sigmoidf_fast(dw[0]  * mw + dw[1]  * mx + dw[2]  * my + dw[3]  * mz + db[0]);
        const float g1 = sigmoidf_fast(dw[4]  * mw + dw[5]  * mx + dw[6]  * my + dw[7]  * mz + db[1]);
        const float g2 = sigmoidf_fast(dw[8]  * mw + dw[9]  * mx + dw[10] * my + dw[11] * mz + db[2]);
        const float g3 = sigmoidf_fast(dw[12] * mw + dw[13] * mx + dw[14] * my + dw[15] * mz + db[3]);
        sw *= g0; sx *= g1; sy *= g2; sz *= g3;

        // ctx += hamilton(spinor, v_j)
        const float4 vq = *(const float4*)(vbase + (size_t)j * EDIM);
        const float vw = vq.x, vx = vq.y, vy = vq.z, vz = vq.w;
        aw += sw * vw - sx * vx - sy * vy - sz * vz;
        ax += sw * vx + sx * vw + sy * vz - sz * vy;
        ay += sw * vy - sx * vz + sy * vw + sz * vx;
        az += sw * vz + sx * vy - sy * vx + sz * vw;
    }

    // Combine the two j-halves of the wave.
    aw += __shfl_xor(aw, 16);
    ax += __shfl_xor(ax, 16);
    ay += __shfl_xor(ay, 16);
    az += __shfl_xor(az, 16);

    if (half == 0) {
        float* cp = ctx + ((size_t)(b * LDIM + i) * EDIM) + headOff;
        cp[0] = aw; cp[1] = ax; cp[2] = ay; cp[3] = az;
    }
}

extern "C" void kernel_launch(void* const* d_in, const int* in_sizes, int n_in,
                              void* d_out, int out_size, void* d_ws, size_t ws_size,
                              hipStream_t stream) {
    const float* x     = (const float*)d_in[0];
    const float* Wq    = (const float*)d_in[1];
    const float* Wk    = (const float*)d_in[2];
    const float* Wv    = (const float*)d_in[3];
    const float* Wo    = (const float*)d_in[4];
    const float* dde_w = (const float*)d_in[5];
    const float* dde_b = (const float*)d_in[6];
    float* out = (float*)d_out;

    float* ws = (float*)d_ws;
    const size_t RC = (size_t)ROWS * EDIM;      // 393216 floats per buffer
    float* qws  = ws;
    float* kws  = ws + RC;
    float* vws  = ws + 2 * RC;
    float* ctxw = ws + 3 * RC;                  // total 6 MB of d_ws

    const dim3 blk(256);
    // GEMM: (768/32)*(512/32) = 384 waves -> 48 blocks of 8 waves.
    const int gemmWaves = (ROWS / 32) * (EDIM / 32);
    const dim3 gemmGrid((gemmWaves * 32 + 255) / 256);

    quat_proj_gemm<<<gemmGrid, blk, 0, stream>>>(x, Wq, qws);
    quat_proj_gemm<<<gemmGrid, blk, 0, stream>>>(x, Wk, kws);
    quat_proj_gemm<<<gemmGrid, blk, 0, stream>>>(x, Wv, vws);

    // Attention: 16*384 = 6144 waves -> 768 blocks of 8 waves.
    const int attnWaves = 16 * LDIM;
    const dim3 attnGrid((attnWaves * 32 + 255) / 256);
    quat_attn<<<attnGrid, blk, 0, stream>>>(qws, kws, vws, dde_w, dde_b, ctxw);

    quat_proj_gemm<<<gemmGrid, blk, 0, stream>>>(ctxw, Wo, out);
}